// Model_87677462380844
// MI455X (gfx1250) — compile-verified
//
#include <hip/hip_runtime.h>
#include <hip/hip_bf16.h>

// MI455X (gfx1250) — memory-bound few-shot scoring kernel.
// 1500 independent problems; each: [512 x 36] train/test blocks.
// fp32 WMMA (v_wmma_f32_16x16x4_f32) used for the top-K fuse matvec.

typedef __attribute__((ext_vector_type(2))) float v2f;
typedef __attribute__((ext_vector_type(8))) float v8f;

#define D_DIM 512
#define P_DIM 36
#define SCALE_CLS 7.0f
#define EPS_N 1e-12f

__global__ __launch_bounds__(256) void fewshot_topk_kernel(
    const float* __restrict__ ftrain,
    const float* __restrict__ ftest,
    const int* __restrict__ Kp,
    float* __restrict__ out,
    int nprob)
{
    const int idx = blockIdx.x;
    const int K = *Kp;
    const size_t base = (size_t)idx * (size_t)(D_DIM * P_DIM);
    const float* __restrict__ fr = ftrain + base;  // fr[d*36 + p]
    const float* __restrict__ ft = ftest  + base;
    const int t = threadIdx.x;

    __shared__ float s_mt[D_DIM], s_mr[D_DIM];     // normalized mean vectors
    __shared__ float s_ft[D_DIM], s_fr[D_DIM];     // fuse vectors
    __shared__ float s_invt[P_DIM], s_invr[P_DIM]; // 1/max(||row||,eps)
    __shared__ float s_St[P_DIM],  s_Sr[P_DIM];    // per-position scores
    __shared__ float s_wt[P_DIM],  s_wr[P_DIM];    // top-K weights
    __shared__ float s_red[2];

    // ---- Phase 1: per-position L2 norms (coalesced: 36 lanes, stride-1 in p)
    if (t < P_DIM) {
        float ar = 0.f, at = 0.f;
        for (int d = 0; d < D_DIM; ++d) {
            float vr = fr[d * P_DIM + t];
            float vt = ft[d * P_DIM + t];
            ar = fmaf(vr, vr, ar);
            at = fmaf(vt, vt, at);
        }
        s_invr[t] = 1.0f / fmaxf(sqrtf(ar), EPS_N);
        s_invt[t] = 1.0f / fmaxf(sqrtf(at), EPS_N);
    }
    // ---- Phase 2: spatial means over p for each d
    for (int d = t; d < D_DIM; d += 256) {
        float sr = 0.f, st = 0.f;
        const float* rr = fr + d * P_DIM;
        const float* rt = ft + d * P_DIM;
        for (int p = 0; p < P_DIM; ++p) { sr += rr[p]; st += rt[p]; }
        s_mr[d] = sr * (1.0f / (float)P_DIM);
        s_mt[d] = st * (1.0f / (float)P_DIM);
    }
    __syncthreads();

    // ---- Global score: 7 * <mt_n, mr_n>
    if (t == 0) {
        float n2t = 0.f, n2r = 0.f, dot = 0.f;
        for (int d = 0; d < D_DIM; ++d) {
            n2t = fmaf(s_mt[d], s_mt[d], n2t);
            n2r = fmaf(s_mr[d], s_mr[d], n2r);
            dot = fmaf(s_mt[d], s_mr[d], dot);
        }
        float it = 1.0f / fmaxf(sqrtf(n2t), EPS_N);
        float ir = 1.0f / fmaxf(sqrtf(n2r), EPS_N);
        s_red[0] = it; s_red[1] = ir;
        out[idx] = SCALE_CLS * dot * it * ir;
    }
    __syncthreads();
    {   // normalize the mean vectors in LDS
        float it = s_red[0], ir = s_red[1];
        for (int d = t; d < D_DIM; d += 256) { s_mt[d] *= it; s_mr[d] *= ir; }
    }
    __syncthreads();

    // ---- Phase 3: per-position scores (coalesced over p; LDS broadcast of mean)
    if (t < P_DIM) {
        float accr = 0.f, acct = 0.f;
        for (int d = 0; d < D_DIM; ++d) {
            accr = fmaf(fr[d * P_DIM + t], s_mt[d], accr);
            acct = fmaf(ft[d * P_DIM + t], s_mr[d], acct);
        }
        s_Sr[t] = SCALE_CLS * s_invr[t] * accr;
        s_St[t] = SCALE_CLS * s_invt[t] * acct;
    }
    __syncthreads();

    // ---- Phase 4: top-K selection via rank counting (jax top_k tie order:
    // strictly greater, or equal value at lower index, outranks us)
    if (t < P_DIM) {
        float vr = s_Sr[t], vt = s_St[t];
        int cr = 0, ct = 0;
        for (int p = 0; p < P_DIM; ++p) {
            float xr = s_Sr[p], xt = s_St[p];
            cr += (xr > vr) || (xr == vr && p < t);
            ct += (xt > vt) || (xt == vt && p < t);
        }
        float invK = 1.0f / (float)K;
        s_wr[t] = (cr < K) ? s_invr[t] * invK : 0.0f;
        s_wt[t] = (ct < K) ? s_invt[t] * invK : 0.0f;
    }
    __syncthreads();

    // ---- Phase 5: fuse[d] = sum_p raw[d][p] * w[p] via fp32 WMMA 16x16x4.
    // A tile: 16 d-rows x 4 p's. ISA layout: lanes 0-15 hold K=0,1 (vgpr0,1),
    // lanes 16-31 hold K=2,3 -> per-lane float2 at raw[(d0+lrow)*36 + k + 2*lh].
    // B: weights broadcast across N (value depends only on K).
    const int lane = t & 31;
    const int lrow = lane & 15;
    const int lh   = lane >> 4;     // 0 for lanes 0-15, 1 for lanes 16-31
    const int wv   = t >> 5;        // wave id 0..7
    for (int tile = wv; tile < (D_DIM / 16); tile += 8) {
        const int d0 = tile * 16;
        const float* apr = fr + (size_t)(d0 + lrow) * P_DIM + lh * 2;
        const float* apt = ft + (size_t)(d0 + lrow) * P_DIM + lh * 2;
        v8f cr = {0.f,0.f,0.f,0.f,0.f,0.f,0.f,0.f};
        v8f ct = {0.f,0.f,0.f,0.f,0.f,0.f,0.f,0.f};
        #pragma unroll
        for (int kk = 0; kk < (P_DIM / 4); ++kk) {   // K = 36 = 9 x 4
            const int kb = kk * 4;
            v2f ar = *(const v2f*)(apr + kb);        // 8B-aligned global load
            v2f at = *(const v2f*)(apt + kb);
            v2f br; br.x = s_wr[kb + lh * 2]; br.y = s_wr[kb + lh * 2 + 1];
            v2f bt; bt.x = s_wt[kb + lh * 2]; bt.y = s_wt[kb + lh * 2 + 1];
            cr = __builtin_amdgcn_wmma_f32_16x16x4_f32(
                     false, ar, false, br, (short)0, cr, false, false);
            ct = __builtin_amdgcn_wmma_f32_16x16x4_f32(
                     false, at, false, bt, (short)0, ct, false, false);
        }
        // Column N=0 of D lives in lane 0 (M=0..7) and lane 16 (M=8..15)
        if (lrow == 0) {
            #pragma unroll
            for (int i = 0; i < 8; ++i) {
                s_fr[d0 + lh * 8 + i] = cr[i];
                s_ft[d0 + lh * 8 + i] = ct[i];
            }
        }
    }
    __syncthreads();

    // ---- Phase 6: normalize fuse vectors, final score
    if (t == 0) {
        float n2t = 0.f, n2r = 0.f, dot = 0.f;
        for (int d = 0; d < D_DIM; ++d) {
            n2t = fmaf(s_ft[d], s_ft[d], n2t);
            n2r = fmaf(s_fr[d], s_fr[d], n2r);
            dot = fmaf(s_ft[d], s_fr[d], dot);
        }
        out[nprob + idx] = SCALE_CLS * dot /
            (fmaxf(sqrtf(n2t), EPS_N) * fmaxf(sqrtf(n2r), EPS_N));
    }
}

extern "C" void kernel_launch(void* const* d_in, const int* in_sizes, int n_in,
                              void* d_out, int out_size, void* d_ws, size_t ws_size,
                              hipStream_t stream) {
    (void)n_in; (void)out_size; (void)d_ws; (void)ws_size;
    const float* ftrain = (const float*)d_in[0];
    const float* ftest  = (const float*)d_in[1];
    const int*   Kp     = (const int*)d_in[2];
    const int nprob = in_sizes[0] / (D_DIM * P_DIM);   // 1500
    fewshot_topk_kernel<<<nprob, 256, 0, stream>>>(ftrain, ftest, Kp,
                                                   (float*)d_out, nprob);
}